// HeteDP_37134287241399
// MI455X (gfx1250) — compile-verified
//
#include <hip/hip_runtime.h>
#include <hip/hip_bf16.h>

// ---------------------------------------------------------------------------
// HAN-like heterogeneous GAT pipeline for MI455X (gfx1250, wave32, WMMA).
// Sizes (fixed by the reference):
//   N=10000, M=3 metapaths, E=320000 edges each, IN_DIM=128, H=8, D=64,
//   F_OUT=512, OUT_DIM=64, NUM_P=2.
// Output = concat(out[N*64], atten[N*N], pnorm[N]) as f32.
// ---------------------------------------------------------------------------

#define N_NODES   10000
#define M_PATHS   3
#define E_EDGES   320000
#define IN_DIM    128
#define HEADS     8
#define HDIM      64
#define F_OUT     512
#define OUT_DIM   64

typedef float v2f __attribute__((ext_vector_type(2)));
typedef float v8f __attribute__((ext_vector_type(8)));

// ---------------------------------------------------------------------------
// Generic fill
// ---------------------------------------------------------------------------
__global__ void fill_f32(float* __restrict__ p, float v, long long count) {
    long long i = (long long)blockIdx.x * blockDim.x + threadIdx.x;
    long long stride = (long long)gridDim.x * blockDim.x;
    for (; i < count; i += stride) p[i] = v;
}

// ---------------------------------------------------------------------------
// f32 WMMA GEMM with async-LDS B staging:
//   C[MxN] = A[MxK] * B[KxN] (+ bias[N]), row-major.
// Block = 256 thr (8 waves). Block handles a 128-row x 64-col output region;
// each wave computes one 16x64 tile (4 accumulators, 4 v_wmma per K-step 4).
// B tile (32 K-rows x 64 cols, 8 KB) is staged into LDS per chunk via
// GLOBAL_LOAD_ASYNC_TO_LDS_B128 (ASYNCcnt) + s_wait_asynccnt + barrier,
// then consumed as LDS broadcasts. A fragments come straight from global
// (each lane owns a distinct row; L0-resident across the chunk).
// Requires M%16==0, N%64==0, K%32==0 (true for all GEMMs here:
// 10000x128x512, 10000x512x128, 10000x512x64).
//
// Fragment layouts (wave32 f32 WMMA 16x16x4, per ISA 7.12.2):
//   A 16x4: lanes 0-15 rows; VGPR pair holds K = kh, kh+1, kh=(lane>>4)*2
//   B 4x16: lanes 0-15 cols; VGPR pair holds rows kh, kh+1
//   C/D:    elem i -> row r0 + (lane>>4)*8 + i, col lane&15
// ---------------------------------------------------------------------------
__global__ void gemm_wmma_f32_lds(const float* __restrict__ A,
                                  const float* __restrict__ B,
                                  float* __restrict__ C,
                                  const float* __restrict__ bias, // may be null
                                  int Mt, int Nt, int K,
                                  int lda, int ldb, int ldc) {
    __shared__ float bt[32 * 64];              // 8 KB B tile

    const int wave = threadIdx.x >> 5;
    const int lane = threadIdx.x & 31;
    const int bm   = blockIdx.x / Nt;
    const int bn   = blockIdx.x - bm * Nt;
    const int tm   = bm * 8 + wave;
    const bool has_tile = (tm < Mt);           // wave-uniform: EXEC all-1 inside
    const int r0 = tm * 16;
    const int c0 = bn * 64;

    const int mrow = lane & 15;
    const int kh   = (lane >> 4) << 1;         // 0 or 2
    const int col  = lane & 15;

    v8f acc[4] = {};
    const float* __restrict__ Ap =
        has_tile ? (A + (size_t)(r0 + mrow) * lda) : A;

    for (int kc = 0; kc < K; kc += 32) {
        // ---- stage B[kc..kc+31][c0..c0+63] -> LDS (coalesced, async) ----
#pragma unroll
        for (int gg = 0; gg < 2; ++gg) {
            const int g  = threadIdx.x + gg * 256;   // 512 groups of 4 floats
            const int r  = g >> 4;                    // 0..31
            const int c4 = (g & 15) << 2;             // 0,4,...,60
            unsigned lds_addr = (unsigned)(uintptr_t)(&bt[r * 64 + c4]);
            unsigned long long gaddr =
                (unsigned long long)(uintptr_t)(B + (size_t)(kc + r) * ldb + c0 + c4);
            asm volatile("global_load_async_to_lds_b128 %0, %1, off"
                         :: "v"(lds_addr), "v"(gaddr) : "memory");
        }
        asm volatile("s_wait_asynccnt 0" ::: "memory");
        __syncthreads();

        if (has_tile) {
            if (kc + 32 < K) __builtin_prefetch(Ap + kc + 32, 0, 1); // next A chunk
#pragma unroll
            for (int kk = 0; kk < 32; kk += 4) {
                v2f a;
                a.x = Ap[kc + kk + kh];
                a.y = Ap[kc + kk + kh + 1];
                const float* l0 = &bt[(kk + kh) * 64 + col];
                const float* l1 = l0 + 64;
#pragma unroll
                for (int j = 0; j < 4; ++j) {
                    v2f b;
                    b.x = l0[j * 16];
                    b.y = l1[j * 16];
                    acc[j] = __builtin_amdgcn_wmma_f32_16x16x4_f32(
                        false, a, false, b, (short)0, acc[j], false, false);
                }
            }
        }
        __syncthreads();
    }

    if (has_tile) {
        const int rbase = r0 + ((lane >> 4) << 3);
#pragma unroll
        for (int j = 0; j < 4; ++j) {
            const int cc = c0 + j * 16 + col;
            const float bv = bias ? bias[cc] : 0.0f;
#pragma unroll
            for (int i = 0; i < 8; ++i) {
                C[(size_t)(rbase + i) * ldc + cc] = acc[j][i] + bv;
            }
        }
    }
}

// ---------------------------------------------------------------------------
// el/er: per (node, head) dot(feat[n,h,:], attn_{l,r}[h,:]) over D=64
// ---------------------------------------------------------------------------
__global__ void compute_eler(const float* __restrict__ feat,
                             const float* __restrict__ al,
                             const float* __restrict__ ar,
                             float* __restrict__ el, float* __restrict__ er,
                             int NH) {
    int t = blockIdx.x * blockDim.x + threadIdx.x;
    if (t >= NH) return;
    int n = t >> 3, h = t & 7;
    const float* f = feat + (size_t)n * F_OUT + h * HDIM;
    const float* l = al + h * HDIM;
    const float* r = ar + h * HDIM;
    float sl = 0.f, sr = 0.f;
#pragma unroll 8
    for (int d = 0; d < HDIM; ++d) { float fv = f[d]; sl += fv * l[d]; sr += fv * r[d]; }
    el[t] = sl; er[t] = sr;
}

// ---------------------------------------------------------------------------
// float atomic max (sign-aware int/uint ordering trick)
// ---------------------------------------------------------------------------
__device__ __forceinline__ void atomicMaxFloat(float* addr, float val) {
    if (val >= 0.0f) atomicMax((int*)addr, __float_as_int(val));
    else             atomicMin((unsigned int*)addr, __float_as_uint(val));
}

__device__ __forceinline__ float leaky(float x, float s) { return x > 0.f ? x : s * x; }

// segment_max over dst of leaky_relu(el[src]+er[dst], 0.2)
__global__ void edge_max(const float* __restrict__ el, const float* __restrict__ er,
                         const int* __restrict__ src, const int* __restrict__ dst,
                         float* __restrict__ emax, int EH) {
    int t = blockIdx.x * blockDim.x + threadIdx.x;
    if (t >= EH) return;
    int e = t >> 3, h = t & 7;
    int s = src[e], d = dst[e];
    float x = leaky(el[s * HEADS + h] + er[d * HEADS + h], 0.2f);
    atomicMaxFloat(&emax[d * HEADS + h], x);
}

// segment_sum over dst of exp(e - emax[dst])
__global__ void edge_expsum(const float* __restrict__ el, const float* __restrict__ er,
                            const int* __restrict__ src, const int* __restrict__ dst,
                            const float* __restrict__ emax, float* __restrict__ denom,
                            int EH) {
    int t = blockIdx.x * blockDim.x + threadIdx.x;
    if (t >= EH) return;
    int e = t >> 3, h = t & 7;
    int s = src[e], d = dst[e];
    float x = leaky(el[s * HEADS + h] + er[d * HEADS + h], 0.2f);
    atomicAdd(&denom[d * HEADS + h], __expf(x - emax[d * HEADS + h]));
}

// ---------------------------------------------------------------------------
// Aggregation: one block (256 thr) per edge.
//   alpha[h] = exp(e_h - emax)/denom;   rst[dst,:] += alpha[h]*feat[src,:]
//   alpha_mean[e] = mean_h alpha[h]
// ---------------------------------------------------------------------------
__global__ void gat_aggregate(const float* __restrict__ feat,
                              const float* __restrict__ el, const float* __restrict__ er,
                              const float* __restrict__ emax, const float* __restrict__ denom,
                              const int* __restrict__ src, const int* __restrict__ dst,
                              float* __restrict__ rst, float* __restrict__ alpha_mean) {
    __shared__ float s_alpha[HEADS];
    __shared__ int s_sd[2];
    const int e = blockIdx.x;
    if (threadIdx.x == 0) { s_sd[0] = src[e]; s_sd[1] = dst[e]; }
    __syncthreads();
    const int sn = s_sd[0], dn = s_sd[1];
    if (threadIdx.x < HEADS) {
        const int h = threadIdx.x;
        float x = leaky(el[sn * HEADS + h] + er[dn * HEADS + h], 0.2f);
        s_alpha[h] = __expf(x - emax[dn * HEADS + h]) / denom[dn * HEADS + h];
    }
    __syncthreads();
    if (threadIdx.x == 0) {
        float s = 0.f;
#pragma unroll
        for (int h = 0; h < HEADS; ++h) s += s_alpha[h];
        alpha_mean[e] = s * (1.0f / HEADS);
    }
    const float* fs = feat + (size_t)sn * F_OUT;
    float* rd = rst + (size_t)dn * F_OUT;
    for (int idx = threadIdx.x; idx < F_OUT; idx += blockDim.x) {
        atomicAdd(&rd[idx], s_alpha[idx >> 6] * fs[idx]);
    }
}

// zp = elu(rst + gat_bias)
__global__ void finalize_zp(float* __restrict__ zp, const float* __restrict__ bias,
                            int total) {
    int t = blockIdx.x * blockDim.x + threadIdx.x;
    if (t >= total) return;
    float v = zp[t] + bias[t & (F_OUT - 1)];
    zp[t] = v > 0.f ? v : (__expf(v) - 1.0f);
}

// wsum[m] += mean_n leaky((y[n,:]+sa_b1) . sa_w2, 0.01)
__global__ void semantic_reduce(const float* __restrict__ y,
                                const float* __restrict__ sa_b1,
                                const float* __restrict__ sa_w2,
                                float* __restrict__ wsum, int n_nodes, float invN) {
    int n = blockIdx.x * blockDim.x + threadIdx.x;
    if (n >= n_nodes) return;
    const float* yr = y + (size_t)n * 128;
    float s = 0.f;
#pragma unroll 8
    for (int j = 0; j < 128; ++j) s += (yr[j] + sa_b1[j]) * sa_w2[j];
    s = leaky(s, 0.01f);
    atomicAdd(wsum, s * invN);
}

// beta = softmax(wsum) over M=3
__global__ void compute_beta(const float* __restrict__ wsum, float* __restrict__ beta) {
    float w0 = wsum[0], w1 = wsum[1], w2 = wsum[2];
    float mx = fmaxf(w0, fmaxf(w1, w2));
    float e0 = __expf(w0 - mx), e1 = __expf(w1 - mx), e2 = __expf(w2 - mx);
    float inv = 1.0f / (e0 + e1 + e2);
    beta[0] = e0 * inv; beta[1] = e1 * inv; beta[2] = e2 * inv;
}

// atten[src,dst] += alpha_mean * beta[m]  (duplicate edges accumulate)
__global__ void atten_scatter(const int* __restrict__ src, const int* __restrict__ dst,
                              const float* __restrict__ alpha_mean,
                              const float* __restrict__ beta,
                              float* __restrict__ atten, int total) {
    int t = blockIdx.x * blockDim.x + threadIdx.x;
    if (t >= total) return;
    int m = t / E_EDGES;
    size_t s = (size_t)src[t];
    size_t d = (size_t)dst[t];
    atomicAdd(&atten[s * (size_t)N_NODES + d], alpha_mean[t] * beta[m]);
}

// z = sum_m beta[m] * zp[m]
__global__ void combine_z(const float* __restrict__ zp, const float* __restrict__ beta,
                          float* __restrict__ z, int total) {
    int t = blockIdx.x * blockDim.x + threadIdx.x;
    if (t >= total) return;
    z[t] = beta[0] * zp[t] + beta[1] * zp[(size_t)total + t] + beta[2] * zp[2 * (size_t)total + t];
}

// per-node MLP + L2 norm: one 64-thread block per node
__global__ void pnorm_kernel(const float* __restrict__ p, const float* __restrict__ w1,
                             const float* __restrict__ w2, const float* __restrict__ b2,
                             float* __restrict__ out) {
    __shared__ float s_a[OUT_DIM];
    __shared__ float s_r[OUT_DIM];
    const int n = blockIdx.x;
    const int j = threadIdx.x;
    float a = p[n * 2 + 0] * w1[0 * OUT_DIM + j] + p[n * 2 + 1] * w1[1 * OUT_DIM + j];
    s_a[j] = leaky(a, 0.2f);
    __syncthreads();
    float b = b2[j];
#pragma unroll 8
    for (int k = 0; k < OUT_DIM; ++k) b += s_a[k] * w2[k * OUT_DIM + j];
    b = leaky(b, 0.01f);
    s_r[j] = b * b;
    __syncthreads();
    for (int off = 32; off > 0; off >>= 1) {
        if (j < off) s_r[j] += s_r[j + off];
        __syncthreads();
    }
    if (j == 0) out[n] = sqrtf(s_r[0]);
}

// ---------------------------------------------------------------------------
// Host-side orchestration
// ---------------------------------------------------------------------------
static inline void launch_gemm(const float* A, const float* B, float* C,
                               const float* bias, int M, int Nc, int K,
                               hipStream_t stream) {
    int Mt = M / 16, Nt = Nc / 64;
    int blocks = ((Mt + 7) / 8) * Nt;          // 8 row-tiles (waves) per block
    gemm_wmma_f32_lds<<<blocks, 256, 0, stream>>>(A, B, C, bias, Mt, Nt, K, K, Nc, Nc);
}

extern "C" void kernel_launch(void* const* d_in, const int* in_sizes, int n_in,
                              void* d_out, int out_size, void* d_ws, size_t ws_size,
                              hipStream_t stream) {
    const float* h        = (const float*)d_in[0];
    const int*   src      = (const int*)  d_in[1];
    const int*   dst      = (const int*)  d_in[2];
    const float* fc_w     = (const float*)d_in[3];
    const float* attn_l   = (const float*)d_in[4];
    const float* attn_r   = (const float*)d_in[5];
    const float* gat_bias = (const float*)d_in[6];
    const float* sa_w1    = (const float*)d_in[7];
    const float* sa_b1    = (const float*)d_in[8];
    const float* sa_w2    = (const float*)d_in[9];
    const float* pw       = (const float*)d_in[10];
    const float* pb       = (const float*)d_in[11];
    const float* mlp_w1   = (const float*)d_in[12];
    const float* mlp_w2   = (const float*)d_in[13];
    const float* mlp_b2   = (const float*)d_in[14];
    const float* p_values = (const float*)d_in[15];

    float* out   = (float*)d_out;                       // [N, 64]
    float* atten = out + (size_t)N_NODES * OUT_DIM;     // [N, N]
    float* pnorm = atten + (size_t)N_NODES * N_NODES;   // [N]

    // ---- workspace layout (floats, all 16B-aligned regions) ----
    const size_t NF = (size_t)N_NODES * F_OUT;          // 5,120,000
    float* ws         = (float*)d_ws;
    float* feat       = ws;                 // 3*NF
    float* zp         = feat + 3 * NF;      // 3*NF   (rst accum, then elu in-place)
    float* el         = zp + 3 * NF;        // N*H
    float* er         = el + (size_t)N_NODES * HEADS;   // N*H
    float* emax       = er + (size_t)N_NODES * HEADS;   // N*H
    float* denom      = emax + (size_t)N_NODES * HEADS; // N*H
    float* alpha_mean = denom + (size_t)N_NODES * HEADS;// 3*E
    float* wsum       = alpha_mean + (size_t)M_PATHS * E_EDGES; // 16 (3 used)
    float* beta       = wsum + 16;                      // 16 (3 used)
    float* zbuf       = beta + 16;                      // NF
    float* ybuf       = zbuf + NF;                      // N*128

    const int NH = N_NODES * HEADS;
    const int EH = E_EDGES * HEADS;

    // ---- global zero-init (ws/d_out are poisoned by harness) ----
    fill_f32<<<4096, 256, 0, stream>>>(zp, 0.0f, (long long)(3 * NF));
    fill_f32<<<1, 32, 0, stream>>>(wsum, 0.0f, M_PATHS);
    fill_f32<<<8192, 256, 0, stream>>>(atten, 0.0f, (long long)N_NODES * N_NODES);

    // ---- per-metapath GAT ----
    for (int m = 0; m < M_PATHS; ++m) {
        const float* W    = fc_w + (size_t)m * IN_DIM * F_OUT;
        const float* al   = attn_l + (size_t)m * HEADS * HDIM;
        const float* ar   = attn_r + (size_t)m * HEADS * HDIM;
        const float* gb   = gat_bias + (size_t)m * F_OUT;
        const int*   sm   = src + (size_t)m * E_EDGES;
        const int*   dm   = dst + (size_t)m * E_EDGES;
        float* feat_m = feat + m * NF;
        float* zp_m   = zp + m * NF;
        float* am_m   = alpha_mean + (size_t)m * E_EDGES;

        // feat = h @ W   (10000x128 @ 128x512)  -> WMMA f32 + async-LDS staging
        launch_gemm(h, W, feat_m, nullptr, N_NODES, F_OUT, IN_DIM, stream);

        // el / er
        compute_eler<<<(NH + 255) / 256, 256, 0, stream>>>(feat_m, al, ar, el, er, NH);

        // edge softmax over dst
        fill_f32<<<(NH + 255) / 256, 256, 0, stream>>>(emax, -INFINITY, NH);
        fill_f32<<<(NH + 255) / 256, 256, 0, stream>>>(denom, 0.0f, NH);
        edge_max<<<(EH + 255) / 256, 256, 0, stream>>>(el, er, sm, dm, emax, EH);
        edge_expsum<<<(EH + 255) / 256, 256, 0, stream>>>(el, er, sm, dm, emax, denom, EH);

        // rst[dst] += alpha * feat[src]  (+ alpha mean over heads)
        gat_aggregate<<<E_EDGES, 256, 0, stream>>>(feat_m, el, er, emax, denom,
                                                   sm, dm, zp_m, am_m);

        // zp = elu(rst + bias)
        finalize_zp<<<(int)((NF + 255) / 256), 256, 0, stream>>>(zp_m, gb, (int)NF);

        // semantic attention: y = zp_m @ sa_w1 (10000x512 @ 512x128) -> WMMA f32
        launch_gemm(zp_m, sa_w1, ybuf, nullptr, N_NODES, 128, F_OUT, stream);
        semantic_reduce<<<(N_NODES + 255) / 256, 256, 0, stream>>>(
            ybuf, sa_b1, sa_w2, wsum + m, N_NODES, 1.0f / N_NODES);
    }

    // beta = softmax(wsum)
    compute_beta<<<1, 1, 0, stream>>>(wsum, beta);

    // dense attention scatter
    atten_scatter<<<(M_PATHS * E_EDGES + 255) / 256, 256, 0, stream>>>(
        src, dst, alpha_mean, beta, atten, M_PATHS * E_EDGES);

    // z = sum_m beta[m]*zp[m] ; out = z @ pw + pb  -> WMMA f32
    combine_z<<<(int)((NF + 255) / 256), 256, 0, stream>>>(zp, beta, zbuf, (int)NF);
    launch_gemm(zbuf, pw, out, pb, N_NODES, OUT_DIM, F_OUT, stream);

    // pnorm
    pnorm_kernel<<<N_NODES, OUT_DIM, 0, stream>>>(p_values, mlp_w1, mlp_w2, mlp_b2, pnorm);
}